// DontCareLoss_74869869904427
// MI455X (gfx1250) — compile-verified
//
#include <hip/hip_runtime.h>

// DontCareLoss, MI455X (gfx1250, wave32).
// Memory-bound streaming reduction: one f32 read of input (164 MB -> ~7us @ 23.3 TB/s).
// Bulk sum(x^2) with float4 loads + sparse per-row corrections, then a
// deterministic two-pass reduction. Intra-wave reduction uses
// V_WMMA_F32_16X16X4_F32 (A = lane partials, B = ones -> row sums).

#define BLOCK 256
#define MAX_K 64

typedef float v2f __attribute__((ext_vector_type(2)));
typedef float v8f __attribute__((ext_vector_type(8)));

// Reduce one wave32's per-lane partials to a single scalar using WMMA.
// A (16x4): lanes 0-15 carry {K=0,K=1}, lanes 16-31 carry {K=2,K=3}; we place
// acc in the first slot and 0 in the second, so A[m][0]=acc[m], A[m][2]=acc[m+16].
// With B = all-ones (4x16): D[i][j] = acc[i] + acc[i+16] for every column j.
// C/D layout: lane L(0-15) VGPR r holds D[r][L]; lane L(16-31) holds D[8+r][L-16].
// So sum(c[0..7]) gives the low-half sum on lanes 0-15 and the high-half sum on
// lanes 16-31; one shfl_xor(16) completes the 32-lane total.
__device__ __forceinline__ float wave_reduce_wmma(float acc) {
    v2f a; a.x = acc;  a.y = 0.0f;
    v2f b; b.x = 1.0f; b.y = 1.0f;
    v8f c = {};
    c = __builtin_amdgcn_wmma_f32_16x16x4_f32(
        /*neg_a=*/false, a, /*neg_b=*/false, b,
        /*c_mod=*/(short)0, c, /*reuse_a=*/false, /*reuse_b=*/false);
    float s = c[0] + c[1] + c[2] + c[3] + c[4] + c[5] + c[6] + c[7];
    s += __shfl_xor(s, 16, 32);
    return s;  // full wave sum in every lane
}

__global__ __launch_bounds__(BLOCK)
void dcl_row_kernel(const float* __restrict__ input,
                    const int* __restrict__ target,
                    const int* __restrict__ dont_care,
                    float* __restrict__ partial,
                    int C, int K) {
    __shared__ int   s_dc[MAX_K];
    __shared__ float s_wave[BLOCK / 32];

    const int n   = blockIdx.x;
    const int tid = threadIdx.x;
    const float* __restrict__ row = input + (size_t)n * (size_t)C;
    const int tgt = target[n];

    if (tid < K) s_dc[tid] = dont_care[(size_t)n * (size_t)K + tid];
    __syncthreads();

    // Bulk: sum of squares over the row, 16B loads per lane.
    float acc = 0.0f;
    const int C4 = C >> 2;
    const float4* __restrict__ row4 = (const float4*)row;
    for (int i = tid; i < C4; i += BLOCK) {
        float4 v = row4[i];
        if (i + 2 * BLOCK < C4)  // guarded speculative prefetch -> global_prefetch_b8
            __builtin_prefetch((const void*)(row4 + i + 2 * BLOCK), 0, 0);
        acc = fmaf(v.x, v.x, acc);
        acc = fmaf(v.y, v.y, acc);
        acc = fmaf(v.z, v.z, acc);
        acc = fmaf(v.w, v.w, acc);
    }
    for (int i = (C4 << 2) + tid; i < C; i += BLOCK) {
        float x = row[i];
        acc = fmaf(x, x, acc);
    }

    // Sparse corrections.
    // Don't-care columns (deduplicated; target branch takes precedence): -x^2.
    if (tid < K) {
        const int j = s_dc[tid];
        bool apply = (j != tgt);
        if (apply) {
            for (int k = 0; k < tid; ++k) {
                if (s_dc[k] == j) { apply = false; break; }
            }
        }
        if (apply) {
            const float x = row[j];
            acc -= x * x;
        }
    }
    // Target column: (1-x)^2 replaces x^2  =>  +(1 - 2x).
    if (tid == MAX_K) {
        const float xt = row[tgt];
        acc += 1.0f - 2.0f * xt;
    }

    // Wave reduce via WMMA (EXEC all-ones here: no divergence at this point),
    // then combine the 8 waves through LDS.
    const float wsum = wave_reduce_wmma(acc);
    const int wave = tid >> 5;
    const int lane = tid & 31;
    if (lane == 0) s_wave[wave] = wsum;
    __syncthreads();
    if (tid == 0) {
        float t = 0.0f;
#pragma unroll
        for (int w = 0; w < BLOCK / 32; ++w) t += s_wave[w];
        partial[n] = t;
    }
}

__global__ __launch_bounds__(BLOCK)
void dcl_final_kernel(const float* __restrict__ partial,
                      float* __restrict__ out, int n) {
    __shared__ float s_wave[BLOCK / 32];
    const int tid = threadIdx.x;

    float acc = 0.0f;
    for (int i = tid; i < n; i += BLOCK) acc += partial[i];

    const float wsum = wave_reduce_wmma(acc);
    const int wave = tid >> 5;
    const int lane = tid & 31;
    if (lane == 0) s_wave[wave] = wsum;
    __syncthreads();
    if (tid == 0) {
        float t = 0.0f;
#pragma unroll
        for (int w = 0; w < BLOCK / 32; ++w) t += s_wave[w];
        out[0] = t;
    }
}

extern "C" void kernel_launch(void* const* d_in, const int* in_sizes, int n_in,
                              void* d_out, int out_size, void* d_ws, size_t ws_size,
                              hipStream_t stream) {
    const float* input     = (const float*)d_in[0];  // [N, C] f32
    const int*   target    = (const int*)d_in[1];    // [N]    int
    const int*   dont_care = (const int*)d_in[2];    // [N, K] int
    float* out     = (float*)d_out;
    float* partial = (float*)d_ws;                   // N floats of scratch

    const int N = in_sizes[1];       // 4096
    const int C = in_sizes[0] / N;   // 10000
    const int K = in_sizes[2] / N;   // 64

    dcl_row_kernel<<<N, BLOCK, 0, stream>>>(input, target, dont_care, partial, C, K);
    dcl_final_kernel<<<1, BLOCK, 0, stream>>>(partial, out, N);
}